// LEConv_18734647345433
// MI455X (gfx1250) — compile-verified
//
#include <hip/hip_runtime.h>

typedef float v2f __attribute__((ext_vector_type(2)));
typedef float v8f __attribute__((ext_vector_type(8)));

#define NN 50000
#define NE 800000
#define CH 128

// ---------------------------------------------------------------- zero deg
__global__ void LEConv_zero_deg(float* __restrict__ deg) {
    int i = blockIdx.x * blockDim.x + threadIdx.x;
    if (i < NN) deg[i] = 0.0f;
}

// ---------------------------------------------------------------- 3 fused GEMMs via f32 WMMA
// H   = x @ W            (weight is [K=128][N=128], row-major K-major)
// L1  = x @ W1^T         (lin1_w is [N=128][K=128] -> B[k][n] = W1[n*128+k], contiguous in k)
// out = x @ W2^T + b2    (written directly to d_out; edge atomics land on top)
//
// The 16x128 x-tile is staged into LDS once per block via the CDNA5 async
// global->LDS path (ASYNCcnt), then A-fragments come from LDS (ds_load_b64).
__global__ __launch_bounds__(256) void LEConv_gemm3_wmma(
    const float* __restrict__ x,
    const float* __restrict__ W,
    const float* __restrict__ W1,
    const float* __restrict__ W2,
    const float* __restrict__ b2,
    float* __restrict__ H,
    float* __restrict__ L1,
    float* __restrict__ out)
{
    __shared__ float xs[16 * CH];              // 8 KB x-tile, shared by all 8 waves

    const int wave  = threadIdx.x >> 5;        // 8 waves -> N tiles 0..7
    const int lane  = threadIdx.x & 31;
    const int m0    = blockIdx.x * 16;         // 3125 blocks * 16 rows = 50000
    const int n0    = wave * 16;

    // --- async stage of the x tile: 2048 floats, 8 per thread, 2 x B128 each
    {
        const unsigned ldsa = (unsigned)(uintptr_t)(&xs[threadIdx.x * 8]);
        const float* gsrc   = x + (size_t)m0 * CH + threadIdx.x * 8;
        asm volatile(
            "global_load_async_to_lds_b128 %0, %1, off\n\t"
            "global_load_async_to_lds_b128 %0, %1, off offset:16\n\t"
            "s_wait_asynccnt 0x0"
            :
            : "v"(ldsa), "v"(gsrc)
            : "memory");
    }
    __syncthreads();

    // A fragment mapping (16x4 f32): lanes 0-15 -> M=lane, VGPR{0,1} = K{k,k+1};
    //                                lanes 16-31 -> M=lane-16, VGPR{0,1} = K{k+2,k+3}
    const int arow  = lane & 15;
    const int khalf = (lane >> 4) * 2;
    // B fragment mapping (4x16 f32): lanes 0-15 -> N=n0+lane, rows K{k,k+1};
    //                                lanes 16-31 -> N=n0+lane-16, rows K{k+2,k+3}
    const int ncol  = n0 + (lane & 15);

    v8f acc0 = {};   // x @ W
    v8f acc1 = {};   // x @ W1^T
    v8f acc2 = {};   // x @ W2^T

    const float* xsrow = &xs[arow * CH];

    for (int k0 = 0; k0 < CH; k0 += 4) {
        const int ka = k0 + khalf;

        v2f a = *(const v2f*)(xsrow + ka);         // ds_load_b64 from LDS

        v2f bw;                                    // W: stride-CH column reads (L2-hot)
        bw.x = W[(size_t)ka * CH + ncol];
        bw.y = W[(size_t)(ka + 1) * CH + ncol];

        v2f bw1 = *(const v2f*)(W1 + (size_t)ncol * CH + ka);   // transposed: contiguous
        v2f bw2 = *(const v2f*)(W2 + (size_t)ncol * CH + ka);

        acc0 = __builtin_amdgcn_wmma_f32_16x16x4_f32(false, a, false, bw,
                                                     (short)0, acc0, false, false);
        acc1 = __builtin_amdgcn_wmma_f32_16x16x4_f32(false, a, false, bw1,
                                                     (short)0, acc1, false, false);
        acc2 = __builtin_amdgcn_wmma_f32_16x16x4_f32(false, a, false, bw2,
                                                     (short)0, acc2, false, false);
    }

    // C/D layout: VGPR j -> M = m0 + j (lanes 0-15) or m0 + 8 + j (lanes 16-31), N = ncol
    const float bias2 = b2[ncol];
    const int mbase = m0 + ((lane >> 4) ? 8 : 0);
#pragma unroll
    for (int j = 0; j < 8; ++j) {
        const size_t idx = (size_t)(mbase + j) * CH + ncol;
        H[idx]   = acc0[j];
        L1[idx]  = acc1[j];
        out[idx] = acc2[j] + bias2;   // seed d_out with x@W2^T + b2
    }
}

// ---------------------------------------------------------------- edge scatter phase
// One wave (32 lanes) per edge; lane owns 4 channels (float4 gather, 4 atomic adds).
__global__ __launch_bounds__(256) void LEConv_edge(
    const long long* __restrict__ ei,     // [2][NE] int64
    const float* __restrict__ ew,
    const float* __restrict__ H,
    float* __restrict__ out,
    float* __restrict__ deg)
{
    const int e    = (int)(((size_t)blockIdx.x * blockDim.x + threadIdx.x) >> 5);
    const int lane = threadIdx.x & 31;
    if (e >= NE) return;

    const long long row = ei[e];
    const long long col = ei[NE + e];
    float w = ew[e];
    if (row == col) w = 0.0f;              // remove_self_loops

    if (lane == 0) atomicAdd(deg + row, w);

    if (w != 0.0f) {
        const float4 hv = *(const float4*)(H + (size_t)col * CH + lane * 4);
        float* o = out + (size_t)row * CH + lane * 4;
        atomicAdd(o + 0, w * hv.x);
        atomicAdd(o + 1, w * hv.y);
        atomicAdd(o + 2, w * hv.z);
        atomicAdd(o + 3, w * hv.w);
    }
}

// ---------------------------------------------------------------- final combine
// out += deg * (L1 + b1)
__global__ void LEConv_combine(const float* __restrict__ L1,
                               const float* __restrict__ b1,
                               const float* __restrict__ deg,
                               float* __restrict__ out)
{
    const int i = blockIdx.x * blockDim.x + threadIdx.x;
    if (i < NN * CH) {
        const int node = i >> 7;
        const int c    = i & (CH - 1);
        out[i] += deg[node] * (L1[i] + b1[c]);
    }
}

// ---------------------------------------------------------------- launcher
extern "C" void kernel_launch(void* const* d_in, const int* in_sizes, int n_in,
                              void* d_out, int out_size, void* d_ws, size_t ws_size,
                              hipStream_t stream) {
    const float*     x   = (const float*)d_in[0];
    const long long* ei  = (const long long*)d_in[1];   // int64 edge_index [2][NE]
    const float*     ew  = (const float*)d_in[2];
    const float*     W   = (const float*)d_in[3];       // [128][128]
    const float*     W1  = (const float*)d_in[4];       // lin1_w [128][128]
    const float*     B1  = (const float*)d_in[5];
    const float*     W2  = (const float*)d_in[6];       // lin2_w [128][128]
    const float*     B2  = (const float*)d_in[7];
    float* out = (float*)d_out;

    // workspace layout: H [NN*CH] | L1 [NN*CH] | deg [NN]   (~51.4 MB)
    float* H   = (float*)d_ws;
    float* L1  = H  + (size_t)NN * CH;
    float* deg = L1 + (size_t)NN * CH;

    LEConv_zero_deg<<<(NN + 255) / 256, 256, 0, stream>>>(deg);

    LEConv_gemm3_wmma<<<NN / 16, 256, 0, stream>>>(x, W, W1, W2, B2, H, L1, out);

    const long long edge_threads = (long long)NE * 32;
    LEConv_edge<<<(int)((edge_threads + 255) / 256), 256, 0, stream>>>(ei, ew, H, out, deg);

    LEConv_combine<<<(NN * CH + 255) / 256, 256, 0, stream>>>(L1, B1, deg, out);
}